// GINNet_15719580303915
// MI455X (gfx1250) — compile-verified
//
#include <hip/hip_runtime.h>
#include <math.h>

#define N_NODES   100000
#define N_EDGES   1600000
#define F_IN      64
#define DIM       16
#define N_CLASSES 10
#define BN_EPS    1e-5f

typedef __attribute__((ext_vector_type(2))) float v2f;
typedef __attribute__((ext_vector_type(8))) float v8f;

// D = A(16x4 f32) * B(4x16 f32) + C(16x16 f32), full-precision fp32 WMMA.
__device__ __forceinline__ v8f wmma_k4(v2f a, v2f b, v8f c) {
    return __builtin_amdgcn_wmma_f32_16x16x4_f32(
        /*neg_a=*/false, a, /*neg_b=*/false, b,
        /*c_mod=*/(short)0, c, /*reuse_a=*/false, /*reuse_b=*/false);
}

// ---------------------------------------------------------------------------
// Kernel 1: y[N,16] = x[N,64] @ w1a[64,16]   (project BEFORE aggregation:
// segment_sum is linear, so aggregating in 16-dim space cuts edge traffic 4x)
// One wave (32 lanes) per 16-node tile; 16 chained K=4 WMMAs.
// ---------------------------------------------------------------------------
__global__ void gin_proj64(const float* __restrict__ x,
                           const float* __restrict__ w1a,
                           float* __restrict__ y) {
    const int tile = blockIdx.x;
    const int lane = threadIdx.x;
    const int half = lane >> 4;   // 0: K={k,k+1}, 1: K={k+2,k+3}
    const int l    = lane & 15;
    const int arow = tile * 16 + l;

    v8f c;
    #pragma unroll
    for (int r = 0; r < 8; ++r) c[r] = 0.0f;

    #pragma unroll
    for (int k0 = 0; k0 < F_IN; k0 += 4) {
        const int ka = k0 + half * 2;
        v2f a = *(const v2f*)(x + arow * F_IN + ka);      // 8B aligned
        v2f b;
        b.x = w1a[ka * DIM + l];
        b.y = w1a[ka * DIM + DIM + l];
        c = wmma_k4(a, b, c);
    }
    #pragma unroll
    for (int r = 0; r < 8; ++r)
        y[(tile * 16 + half * 8 + r) * DIM + l] = c[r];
}

// ---------------------------------------------------------------------------
// Kernel 2: agg[dst] += feat[src]  over 16 features.
// 16 consecutive lanes per edge -> coalesced 64B gathers; fp32 atomics land
// in L2 (agg working set = 6.4MB << 192MB L2).
// ---------------------------------------------------------------------------
__global__ void gin_scatter_add(const float* __restrict__ feat,
                                const int* __restrict__ src,
                                const int* __restrict__ dst,
                                float* __restrict__ agg) {
    const int t = blockIdx.x * blockDim.x + threadIdx.x;
    if (t >= N_EDGES * DIM) return;
    const int e = t >> 4;
    const int d = t & 15;
    const int s  = src[e];
    const int dd = dst[e];
    atomicAdd(agg + dd * DIM + d, feat[s * DIM + d]);
}

// ---------------------------------------------------------------------------
// Kernel 3: layer-1 tail per 16-node tile:
//   u = BN1( relu( relu(y + agg + b1a) @ w1b + b1b ) )
// A-fragments staged via LDS (padded row = 17 floats, conflict-free).
// ---------------------------------------------------------------------------
__global__ void gin_layer1(const float* __restrict__ y, const float* __restrict__ agg,
                           const float* __restrict__ b1a,
                           const float* __restrict__ w1b, const float* __restrict__ b1b,
                           const float* __restrict__ g,  const float* __restrict__ be,
                           const float* __restrict__ mu, const float* __restrict__ var,
                           float* __restrict__ u) {
    __shared__ float sA[16][17];
    const int tile = blockIdx.x;
    const int lane = threadIdx.x;
    const int half = lane >> 4;
    const int l    = lane & 15;

    const float bl = b1a[l];
    #pragma unroll
    for (int r = 0; r < 8; ++r) {
        const int row = half * 8 + r;
        const int gi  = (tile * 16 + row) * DIM + l;
        sA[row][l] = fmaxf(y[gi] + agg[gi] + bl, 0.0f);
    }
    __syncthreads();

    const float cb = b1b[l];
    v8f c;
    #pragma unroll
    for (int r = 0; r < 8; ++r) c[r] = cb;

    #pragma unroll
    for (int k0 = 0; k0 < DIM; k0 += 4) {
        const int ka = k0 + half * 2;
        v2f a; a.x = sA[l][ka];           a.y = sA[l][ka + 1];
        v2f b; b.x = w1b[ka * DIM + l];   b.y = w1b[ka * DIM + DIM + l];
        c = wmma_k4(a, b, c);
    }

    const float sc = g[l] * rsqrtf(var[l] + BN_EPS);
    const float mm = mu[l], bb = be[l];
    #pragma unroll
    for (int r = 0; r < 8; ++r) {
        const float t = fmaxf(c[r], 0.0f);
        u[(tile * 16 + half * 8 + r) * DIM + l] = (t - mm) * sc + bb;
    }
}

// ---------------------------------------------------------------------------
// Kernel 4: layer-2 + classifier per 16-node tile:
//   h = BN2( relu( relu((u + agg2) @ w2a + b2a) @ w2b + b2b ) )
//   logits = h @ fc_w + fc_b   (fc B-tile zero-padded 16x10 -> 16x16)
// ---------------------------------------------------------------------------
__global__ void gin_layer2_fc(const float* __restrict__ u,  const float* __restrict__ agg2,
                              const float* __restrict__ w2a, const float* __restrict__ b2a,
                              const float* __restrict__ w2b, const float* __restrict__ b2b,
                              const float* __restrict__ g,   const float* __restrict__ be,
                              const float* __restrict__ mu,  const float* __restrict__ var,
                              const float* __restrict__ fcw, const float* __restrict__ fcb,
                              float* __restrict__ logits) {
    __shared__ float sA[16][17];
    const int tile = blockIdx.x;
    const int lane = threadIdx.x;
    const int half = lane >> 4;
    const int l    = lane & 15;

    // stage 0: s = u + agg2
    #pragma unroll
    for (int r = 0; r < 8; ++r) {
        const int row = half * 8 + r;
        const int gi  = (tile * 16 + row) * DIM + l;
        sA[row][l] = u[gi] + agg2[gi];
    }
    __syncthreads();

    // GEMM 1: s @ w2a + b2a
    v8f c;
    {
        const float cb = b2a[l];
        #pragma unroll
        for (int r = 0; r < 8; ++r) c[r] = cb;
    }
    #pragma unroll
    for (int k0 = 0; k0 < DIM; k0 += 4) {
        const int ka = k0 + half * 2;
        v2f a; a.x = sA[l][ka];          a.y = sA[l][ka + 1];
        v2f b; b.x = w2a[ka * DIM + l];  b.y = w2a[ka * DIM + DIM + l];
        c = wmma_k4(a, b, c);
    }
    __syncthreads();
    #pragma unroll
    for (int r = 0; r < 8; ++r) sA[half * 8 + r][l] = fmaxf(c[r], 0.0f);
    __syncthreads();

    // GEMM 2: relu(.) @ w2b + b2b
    {
        const float cb = b2b[l];
        #pragma unroll
        for (int r = 0; r < 8; ++r) c[r] = cb;
    }
    #pragma unroll
    for (int k0 = 0; k0 < DIM; k0 += 4) {
        const int ka = k0 + half * 2;
        v2f a; a.x = sA[l][ka];          a.y = sA[l][ka + 1];
        v2f b; b.x = w2b[ka * DIM + l];  b.y = w2b[ka * DIM + DIM + l];
        c = wmma_k4(a, b, c);
    }
    __syncthreads();

    // relu + BN2 -> LDS
    {
        const float sc = g[l] * rsqrtf(var[l] + BN_EPS);
        const float mm = mu[l], bb = be[l];
        #pragma unroll
        for (int r = 0; r < 8; ++r) {
            const float t = fmaxf(c[r], 0.0f);
            sA[half * 8 + r][l] = (t - mm) * sc + bb;
        }
    }
    __syncthreads();

    // GEMM 3: h @ fc_w + fc_b  (columns 10..15 padded with zeros)
    {
        const float cb = (l < N_CLASSES) ? fcb[l] : 0.0f;
        #pragma unroll
        for (int r = 0; r < 8; ++r) c[r] = cb;
    }
    #pragma unroll
    for (int k0 = 0; k0 < DIM; k0 += 4) {
        const int ka = k0 + half * 2;
        v2f a; a.x = sA[l][ka];  a.y = sA[l][ka + 1];
        v2f b;
        b.x = (l < N_CLASSES) ? fcw[ka * N_CLASSES + l]       : 0.0f;
        b.y = (l < N_CLASSES) ? fcw[(ka + 1) * N_CLASSES + l] : 0.0f;
        c = wmma_k4(a, b, c);
    }
    if (l < N_CLASSES) {
        #pragma unroll
        for (int r = 0; r < 8; ++r)
            logits[(tile * 16 + half * 8 + r) * N_CLASSES + l] = c[r];
    }
}

// ---------------------------------------------------------------------------
// Kernel 5: in-place log_softmax over the 10 classes, one thread per node.
// ---------------------------------------------------------------------------
__global__ void gin_logsoftmax(float* __restrict__ o) {
    const int i = blockIdx.x * blockDim.x + threadIdx.x;
    if (i >= N_NODES) return;
    float* p = o + (size_t)i * N_CLASSES;
    float v[N_CLASSES];
    float m = -INFINITY;
    #pragma unroll
    for (int j = 0; j < N_CLASSES; ++j) { v[j] = p[j]; m = fmaxf(m, v[j]); }
    float s = 0.0f;
    #pragma unroll
    for (int j = 0; j < N_CLASSES; ++j) s += expf(v[j] - m);
    const float ls = m + logf(s);
    #pragma unroll
    for (int j = 0; j < N_CLASSES; ++j) p[j] = v[j] - ls;
}

__global__ void gin_zero(float* __restrict__ p, int n) {
    const int i = blockIdx.x * blockDim.x + threadIdx.x;
    if (i < n) p[i] = 0.0f;
}

// ---------------------------------------------------------------------------
extern "C" void kernel_launch(void* const* d_in, const int* in_sizes, int n_in,
                              void* d_out, int out_size, void* d_ws, size_t ws_size,
                              hipStream_t stream) {
    (void)in_sizes; (void)n_in; (void)out_size; (void)ws_size;

    const float* x   = (const float*)d_in[0];
    const int*   ei  = (const int*)d_in[1];
    const int*   src = ei;
    const int*   dst = ei + N_EDGES;
    const float* w1a = (const float*)d_in[2];  const float* b1a = (const float*)d_in[3];
    const float* w1b = (const float*)d_in[4];  const float* b1b = (const float*)d_in[5];
    const float* g1  = (const float*)d_in[6];  const float* be1 = (const float*)d_in[7];
    const float* m1  = (const float*)d_in[8];  const float* v1  = (const float*)d_in[9];
    const float* w2a = (const float*)d_in[10]; const float* b2a = (const float*)d_in[11];
    const float* w2b = (const float*)d_in[12]; const float* b2b = (const float*)d_in[13];
    const float* g2  = (const float*)d_in[14]; const float* be2 = (const float*)d_in[15];
    const float* m2  = (const float*)d_in[16]; const float* v2  = (const float*)d_in[17];
    const float* fcw = (const float*)d_in[18]; const float* fcb = (const float*)d_in[19];
    float* out = (float*)d_out;

    const int NF = N_NODES * DIM;              // 1.6M floats per node buffer
    float* ws   = (float*)d_ws;
    float* y    = ws;                          // [N,16] projected layer-1 features
    float* u    = ws + (size_t)NF;             // [N,16] layer-1 output
    float* agg1 = ws + (size_t)2 * NF;         // [N,16] (agg1|agg2 contiguous)
    float* agg2 = ws + (size_t)3 * NF;

    const int TILES = N_NODES / 16;            // 6250, exact
    const int ETHREADS = N_EDGES * DIM;        // 25.6M

    gin_zero<<<(2 * NF + 255) / 256, 256, 0, stream>>>(agg1, 2 * NF);
    gin_proj64<<<TILES, 32, 0, stream>>>(x, w1a, y);
    gin_scatter_add<<<(ETHREADS + 255) / 256, 256, 0, stream>>>(y, src, dst, agg1);
    gin_layer1<<<TILES, 32, 0, stream>>>(y, agg1, b1a, w1b, b1b, g1, be1, m1, v1, u);
    gin_scatter_add<<<(ETHREADS + 255) / 256, 256, 0, stream>>>(u, src, dst, agg2);
    gin_layer2_fc<<<TILES, 32, 0, stream>>>(u, agg2, w2a, b2a, w2b, b2b,
                                            g2, be2, m2, v2, fcw, fcb, out);
    gin_logsoftmax<<<(N_NODES + 255) / 256, 256, 0, stream>>>(out);
}